// SpGraphAttentionLayer_27608049778910
// MI455X (gfx1250) — compile-verified
//
#include <hip/hip_runtime.h>
#include <hip/hip_bf16.h>

#define N_NODES 100000
#define E_EDGES 3200000
#define IN_F    128
#define ATT     128
#define HEADS   8
#define DK      16
#define SLOPE   0.2f
#define NH      (N_NODES * HEADS)

typedef __attribute__((ext_vector_type(2))) float v2f;
typedef __attribute__((ext_vector_type(8))) float v8f;

// ---------- monotone float<->uint ordering key (for atomic segment-max) ----
__device__ __forceinline__ unsigned fkey(float f) {
    unsigned b = __float_as_uint(f);
    return (b & 0x80000000u) ? ~b : (b | 0x80000000u);
}
__device__ __forceinline__ float funkey(unsigned k) {
    unsigned b = (k & 0x80000000u) ? (k ^ 0x80000000u) : ~k;
    return __uint_as_float(b);
}

// ---------- init node tables: smax keys = key(-inf floor) = 0, denom = 0 ----
__global__ void init_tables_kernel(unsigned* __restrict__ smax,
                                   float* __restrict__ denom) {
    int i = blockIdx.x * blockDim.x + threadIdx.x;
    if (i < NH) { smax[i] = 0u; denom[i] = 0.0f; }
}

// ---------- fused GEMM (wx = x@W, f32 WMMA 16x16x4) + per-head projections --
// grid = N/16 blocks of 256 threads (8 waves). Wave w computes the 16x16 tile
// covering rows [16*blk, 16*blk+16) and columns [16*w, 16*w+16) == head w.
__global__ void __launch_bounds__(256)
gat_gemm_kernel(const float* __restrict__ x, const float* __restrict__ W,
                const float* __restrict__ a, float* __restrict__ wx,
                float* __restrict__ s_src, float* __restrict__ s_dst) {
    __shared__ float Wl[IN_F * ATT];          // 64 KB of 320 KB/WGP

    const int tid = threadIdx.x;
    // stage W into LDS (float4, 16 per thread)
    {
        const float4* Wg4 = (const float4*)W;
        float4*       Wl4 = (float4*)Wl;
        #pragma unroll
        for (int i = tid; i < (IN_F * ATT) / 4; i += 256) Wl4[i] = Wg4[i];
    }
    __syncthreads();

    const int wave = tid >> 5;                // == head index (0..7)
    const int lane = tid & 31;
    const int half = lane >> 4;               // 0: lanes 0-15, 1: lanes 16-31
    const int l    = lane & 15;
    const int Mbase = blockIdx.x * 16;        // 100000 = 6250*16, exact
    const int row   = Mbase + l;              // A-fragment row for this lane
    const int col   = wave * 16 + l;          // B/C-fragment column

    const float* xr = x + (size_t)row * IN_F + 2 * half;

    v8f acc = {};
    #pragma unroll
    for (int kb = 0; kb < IN_F; kb += 4) {
        // A 16x4 f32 layout: v0 = K(2*half), v1 = K(2*half+1) -> one float2 load
        float2 t = *(const float2*)(xr + kb);
        v2f af; af.x = t.x; af.y = t.y;
        // B 4x16 f32 layout: VGPR j, lanes 0-15 -> K=j, lanes 16-31 -> K=j+2
        v2f bf;
        bf.x = Wl[(kb + 2 * half) * ATT + col];
        bf.y = Wl[(kb + 2 * half + 1) * ATT + col];
        acc = __builtin_amdgcn_wmma_f32_16x16x4_f32(
            /*neg_a=*/false, af, /*neg_b=*/false, bf,
            /*c_mod=*/(short)0, acc, /*reuse_a=*/false, /*reuse_b=*/false);
    }

    // epilogue: store wx tile, and fold the head-w 16-dim dot with a1/a2
    const float a1v = a[l];
    const float a2v = a[DK + l];
    #pragma unroll
    for (int r = 0; r < 8; r++) {
        const int orow = Mbase + r + 8 * half;   // C layout: lanes16-31 = M=r+8
        const float v = acc[r];
        wx[(size_t)orow * ATT + col] = v;
        float ts = v * a1v;
        float td = v * a2v;
        #pragma unroll
        for (int m = 8; m >= 1; m >>= 1) {       // reduce within 16-lane half
            ts += __shfl_xor(ts, m, 32);
            td += __shfl_xor(td, m, 32);
        }
        if (l == 0) {
            s_src[orow * HEADS + wave] = ts;
            s_dst[orow * HEADS + wave] = td;
        }
    }
}

// ---------- per-edge score (recomputed each pass; node tables are L2-hot) ---
__device__ __forceinline__ void edge_scores(const float* s_src, const float* s_dst,
                                            int s, int d, float sc[HEADS]) {
    const float4* ps = (const float4*)(s_src + (size_t)s * HEADS);
    const float4* pd = (const float4*)(s_dst + (size_t)d * HEADS);
    float4 a0 = ps[0], a1 = ps[1], b0 = pd[0], b1 = pd[1];
    sc[0] = a0.x + b0.x; sc[1] = a0.y + b0.y; sc[2] = a0.z + b0.z; sc[3] = a0.w + b0.w;
    sc[4] = a1.x + b1.x; sc[5] = a1.y + b1.y; sc[6] = a1.z + b1.z; sc[7] = a1.w + b1.w;
    #pragma unroll
    for (int h = 0; h < HEADS; h++) {
        float v = sc[h];
        sc[h] = v > 0.0f ? v : SLOPE * v;        // leaky relu
    }
}

// pass A: segment max (keyed by src node) via u32 atomic max on ordered keys
__global__ void edge_max_kernel(const int* __restrict__ edge,
                                const float* __restrict__ s_src,
                                const float* __restrict__ s_dst,
                                unsigned* __restrict__ smax) {
    int e = blockIdx.x * blockDim.x + threadIdx.x;
    if (e >= E_EDGES) return;
    int s = edge[e];
    int d = edge[E_EDGES + e];
    float sc[HEADS];
    edge_scores(s_src, s_dst, s, d, sc);
    #pragma unroll
    for (int h = 0; h < HEADS; h++)
        atomicMax(&smax[(size_t)s * HEADS + h], fkey(sc[h]));
}

// pass B: ex = exp(score - max), atomic segment sum, store ex to output
__global__ void edge_exp_kernel(const int* __restrict__ edge,
                                const float* __restrict__ s_src,
                                const float* __restrict__ s_dst,
                                const unsigned* __restrict__ smax,
                                float* __restrict__ denom,
                                float* __restrict__ att) {
    int e = blockIdx.x * blockDim.x + threadIdx.x;
    if (e >= E_EDGES) return;
    int s = edge[e];
    int d = edge[E_EDGES + e];
    float sc[HEADS];
    edge_scores(s_src, s_dst, s, d, sc);
    float ex[HEADS];
    #pragma unroll
    for (int h = 0; h < HEADS; h++) {
        float m = funkey(smax[(size_t)s * HEADS + h]);
        ex[h] = __expf(sc[h] - m);
        atomicAdd(&denom[(size_t)s * HEADS + h], ex[h]);
    }
    float4* out = (float4*)(att + (size_t)e * HEADS);
    out[0] = make_float4(ex[0], ex[1], ex[2], ex[3]);
    out[1] = make_float4(ex[4], ex[5], ex[6], ex[7]);
}

// pass C: normalize in place
__global__ void edge_norm_kernel(const int* __restrict__ edge,
                                 const float* __restrict__ denom,
                                 float* __restrict__ att) {
    int e = blockIdx.x * blockDim.x + threadIdx.x;
    if (e >= E_EDGES) return;
    int s = edge[e];
    const float4* dn = (const float4*)(denom + (size_t)s * HEADS);
    float4 d0 = dn[0], d1 = dn[1];
    float4* out = (float4*)(att + (size_t)e * HEADS);
    float4 o0 = out[0], o1 = out[1];
    o0.x /= d0.x; o0.y /= d0.y; o0.z /= d0.z; o0.w /= d0.w;
    o1.x /= d1.x; o1.y /= d1.y; o1.z /= d1.z; o1.w /= d1.w;
    out[0] = o0; out[1] = o1;
}

extern "C" void kernel_launch(void* const* d_in, const int* in_sizes, int n_in,
                              void* d_out, int out_size, void* d_ws, size_t ws_size,
                              hipStream_t stream) {
    const float* x    = (const float*)d_in[0];
    const int*   edge = (const int*)d_in[1];
    const float* W    = (const float*)d_in[2];
    const float* a    = (const float*)d_in[3];

    // outputs: attention (E*HEADS) then wx (N*ATT), both f32
    float* att = (float*)d_out;
    float* wx  = att + (size_t)E_EDGES * HEADS;

    // workspace: s_src | s_dst | smax keys | denom  (4 * 3.2 MB = 12.8 MB)
    float*    s_src = (float*)d_ws;
    float*    s_dst = s_src + NH;
    unsigned* smax  = (unsigned*)(s_dst + NH);
    float*    denom = (float*)(smax + NH);

    init_tables_kernel<<<(NH + 255) / 256, 256, 0, stream>>>(smax, denom);
    gat_gemm_kernel<<<N_NODES / 16, 256, 0, stream>>>(x, W, a, wx, s_src, s_dst);

    const int eblk = (E_EDGES + 255) / 256;
    edge_max_kernel <<<eblk, 256, 0, stream>>>(edge, s_src, s_dst, smax);
    edge_exp_kernel <<<eblk, 256, 0, stream>>>(edge, s_src, s_dst, smax, denom, att);
    edge_norm_kernel<<<eblk, 256, 0, stream>>>(edge, denom, att);
}